// AttentionLayer_11776800325798
// MI455X (gfx1250) — compile-verified
//
#include <hip/hip_runtime.h>
#include <math.h>

// ---- problem constants (from reference) ----
#define N_NODES 50000
#define N_EDGES 500000
#define TT      3
#define RR      5
#define HH      8
#define DKK     16
#define FEAT    128
#define NRSEG   (N_NODES * RR)
#define INV_SQRT_DK 0.25f
#define LN_EPS  1e-5f

typedef float v2f __attribute__((ext_vector_type(2)));
typedef float v8f __attribute__((ext_vector_type(8)));

// ---------------------------------------------------------------------------
// WMMA typed projection: out[n,:] = in[n,:] @ W[type[n]] + b[type[n]]
// Computes all T=3 types per 16x16 tile (like the reference) and selects.
// One wave per (16-node tile, 16-col tile). K=128 swept in 32 steps of
// V_WMMA_F32_16X16X4_F32 (exact f32 math, no precision loss).
// A layout (16x4 f32): lanes 0-15 -> M=0..15, VGPR0/1 = K {0,1}; lanes 16-31 K {2,3}.
// B layout (4x16 f32): lanes 0-15 -> N=0..15, VGPR0/1 = K {0,1}; lanes 16-31 K {2,3}.
// C/D: VGPR j: lanes 0-15 -> M=j, lanes 16-31 -> M=j+8.
// ---------------------------------------------------------------------------
__global__ __launch_bounds__(32)
void typed_proj_wmma(const float* __restrict__ in,    // [N,128]
                     const int*   __restrict__ ntype, // [N]
                     const float* __restrict__ W,     // [T,128,128]
                     const float* __restrict__ b,     // [T,128]
                     float* __restrict__ out)         // [N,128]
{
  const int tile = blockIdx.x;            // node tile (16 rows)
  const int ct   = blockIdx.y;            // column tile (16 cols)
  const int lane = threadIdx.x;           // 0..31
  const int half = lane >> 4;             // 0/1 -> K offset 0/2
  const int l16  = lane & 15;
  const int row  = tile * 16 + l16;       // A-matrix row for this lane
  const int col  = ct * 16 + l16;         // B/C column for this lane

  // Preload this lane's A slices: av[i] = in[row][4*i + 2*half .. +1]
  v2f av[32];
  const float* xr = in + (size_t)row * FEAT + 2 * half;
  #pragma unroll
  for (int i = 0; i < 32; ++i) av[i] = *(const v2f*)(xr + 4 * i);

  v8f acc[TT];
  #pragma unroll
  for (int t = 0; t < TT; ++t) {
    v8f c = {0.f, 0.f, 0.f, 0.f, 0.f, 0.f, 0.f, 0.f};
    const float* Wt = W + (size_t)t * FEAT * FEAT + col;
    #pragma unroll
    for (int i = 0; i < 32; ++i) {
      const int k = 4 * i + 2 * half;
      v2f bb;
      bb.x = Wt[(size_t)k * FEAT];
      bb.y = Wt[(size_t)(k + 1) * FEAT];
      c = __builtin_amdgcn_wmma_f32_16x16x4_f32(false, av[i], false, bb,
                                                (short)0, c, false, false);
    }
    acc[t] = c;
  }

  // Select per-row node type, add bias, store.
  const int rbase = tile * 16 + half * 8;
  #pragma unroll
  for (int j = 0; j < 8; ++j) {
    const int r = rbase + j;
    const int t = ntype[r];
    float v;
    if (t == 0)      v = acc[0][j];
    else if (t == 1) v = acc[1][j];
    else             v = acc[2][j];
    out[(size_t)r * FEAT + col] = v + b[t * FEAT + col];
  }
}

// ---------------------------------------------------------------------------
// Workspace init: segment max = -inf, denom = 0, counts = 0, accumulator = 0
// ---------------------------------------------------------------------------
__global__ void init_ws(float* __restrict__ Mseg, float* __restrict__ Dseg,
                        int* __restrict__ Cnt, float* __restrict__ TN)
{
  const int i = blockIdx.x * blockDim.x + threadIdx.x;
  if (i < NRSEG * HH) { Mseg[i] = -__builtin_huge_valf(); Dseg[i] = 0.f; }
  if (i < NRSEG)        Cnt[i]  = 0;
  if (i < N_NODES * FEAT) TN[i] = 0.f;
}

__device__ inline void atomicMaxF(float* addr, float val)
{
  unsigned int* ua = (unsigned int*)addr;
  unsigned int old = *ua;
  while (true) {
    const float f = __uint_as_float(old);
    if (f >= val) break;
    const unsigned int prev = atomicCAS(ua, old, __float_as_uint(val));
    if (prev == old) break;
    old = prev;
  }
}

// ---------------------------------------------------------------------------
// Edge phase 1: a = (q[dst] . (k[src] @ rel_att[r])) * pri / sqrt(dk)
// plus per-(dst,etype) running max and edge count.
// One thread per (edge, head).
// ---------------------------------------------------------------------------
__global__ __launch_bounds__(256)
void edge_logits(const float* __restrict__ K, const float* __restrict__ Q,
                 const int* __restrict__ src, const int* __restrict__ dst,
                 const int* __restrict__ etype,
                 const float* __restrict__ rel_att,   // [R,H,DK,DK]
                 const float* __restrict__ rel_pri,   // [R,H]
                 float* __restrict__ Abuf,            // [E,H]
                 float* __restrict__ Mseg,            // [NR,H]
                 int* __restrict__ Cnt)               // [NR]
{
  const int idx = blockIdx.x * blockDim.x + threadIdx.x;
  if (idx >= N_EDGES * HH) return;
  const int e = idx >> 3;
  const int h = idx & 7;
  const int s = src[e], d = dst[e], r = etype[e];

  const float* kp = K + (size_t)s * FEAT + h * DKK;
  const float* qp = Q + (size_t)d * FEAT + h * DKK;
  const float* Ar = rel_att + (size_t)(r * HH + h) * DKK * DKK;

  float qv[DKK];
  #pragma unroll
  for (int f = 0; f < DKK; ++f) qv[f] = qp[f];

  float acc = 0.f;
  #pragma unroll
  for (int dd = 0; dd < DKK; ++dd) {
    const float* rowp = Ar + dd * DKK;
    float sdot = 0.f;
    #pragma unroll
    for (int f = 0; f < DKK; ++f) sdot += rowp[f] * qv[f];
    acc += kp[dd] * sdot;
  }
  const float a = acc * rel_pri[r * HH + h] * INV_SQRT_DK;
  Abuf[idx] = a;

  const int seg = d * RR + r;
  atomicMaxF(&Mseg[seg * HH + h], a);
  if (h == 0) atomicAdd(&Cnt[seg], 1);
}

// ---------------------------------------------------------------------------
// Edge phase 2: p = exp(a - segmax), accumulate segment denominator.
// ---------------------------------------------------------------------------
__global__ __launch_bounds__(256)
void edge_exp(const int* __restrict__ dst, const int* __restrict__ etype,
              float* __restrict__ Abuf, const float* __restrict__ Mseg,
              float* __restrict__ Dseg)
{
  const int idx = blockIdx.x * blockDim.x + threadIdx.x;
  if (idx >= N_EDGES * HH) return;
  const int e = idx >> 3;
  const int h = idx & 7;
  const int seg = dst[e] * RR + etype[e];
  const float p = expf(Abuf[idx] - Mseg[seg * HH + h]);
  Abuf[idx] = p;
  atomicAdd(&Dseg[seg * HH + h], p);
}

// ---------------------------------------------------------------------------
// Edge phase 3: attw = p / denom ; msg = v[src] @ rel_msg[r] ;
// accumulate attw*msg directly into the per-dst numerator
// (Σ_r h_seg[n,r] == Σ_{edges into n} attw·val, so no [N,R,H,DK] mailbox).
// ---------------------------------------------------------------------------
__global__ __launch_bounds__(256)
void edge_agg(const float* __restrict__ V, const int* __restrict__ src,
              const int* __restrict__ dst, const int* __restrict__ etype,
              const float* __restrict__ rel_msg,     // [R,H,DK,DK]
              const float* __restrict__ Abuf, const float* __restrict__ Dseg,
              float* __restrict__ TN)                // [N,128]
{
  const int idx = blockIdx.x * blockDim.x + threadIdx.x;
  if (idx >= N_EDGES * HH) return;
  const int e = idx >> 3;
  const int h = idx & 7;
  const int s = src[e], d = dst[e], r = etype[e];
  const int seg = d * RR + r;
  const float attw = Abuf[idx] / Dseg[seg * HH + h];

  const float* vp = V + (size_t)s * FEAT + h * DKK;
  const float* Mr = rel_msg + (size_t)(r * HH + h) * DKK * DKK;

  float val[DKK];
  #pragma unroll
  for (int f = 0; f < DKK; ++f) val[f] = 0.f;
  #pragma unroll
  for (int dd = 0; dd < DKK; ++dd) {
    const float vd = vp[dd];
    const float* rowp = Mr + dd * DKK;
    #pragma unroll
    for (int f = 0; f < DKK; ++f) val[f] += vd * rowp[f];
  }
  float* tp = TN + (size_t)d * FEAT + h * DKK;
  #pragma unroll
  for (int f = 0; f < DKK; ++f) atomicAdd(&tp[f], attw * val[f]);
}

// ---------------------------------------------------------------------------
// Per-node: divide accumulator by number of etypes that delivered messages.
// One wave per node, 4 features per lane.
// ---------------------------------------------------------------------------
__global__ __launch_bounds__(256)
void node_scale(float* __restrict__ TN, const int* __restrict__ Cnt)
{
  const int gtid = blockIdx.x * blockDim.x + threadIdx.x;
  const int node = gtid >> 5;
  const int lane = gtid & 31;
  if (node >= N_NODES) return;
  int active = 0;
  #pragma unroll
  for (int r = 0; r < RR; ++r) active += (Cnt[node * RR + r] > 0) ? 1 : 0;
  const float scale = 1.0f / (float)(active < 1 ? 1 : active);
  float4* tp = (float4*)(TN + (size_t)node * FEAT) + lane;
  float4 t = *tp;
  t.x *= scale; t.y *= scale; t.z *= scale; t.w *= scale;
  *tp = t;
}

// ---------------------------------------------------------------------------
// Final: skip-gated residual + per-type LayerNorm. One wave per node,
// 4 features per lane, shuffle reductions for mean/variance.
// ---------------------------------------------------------------------------
__global__ __launch_bounds__(256)
void node_final(const float* __restrict__ trans, const float* __restrict__ x,
                const int* __restrict__ ntype, const float* __restrict__ skip,
                const float* __restrict__ ln_g, const float* __restrict__ ln_b,
                float* __restrict__ out)
{
  const int gtid = blockIdx.x * blockDim.x + threadIdx.x;
  const int node = gtid >> 5;
  const int lane = gtid & 31;
  if (node >= N_NODES) return;

  const int t = ntype[node];
  const float alpha = 1.0f / (1.0f + expf(-skip[t]));

  const float4 tr = ((const float4*)(trans + (size_t)node * FEAT))[lane];
  const float4 xx = ((const float4*)(x     + (size_t)node * FEAT))[lane];
  float o[4];
  o[0] = tr.x * alpha + xx.x * (1.0f - alpha);
  o[1] = tr.y * alpha + xx.y * (1.0f - alpha);
  o[2] = tr.z * alpha + xx.z * (1.0f - alpha);
  o[3] = tr.w * alpha + xx.w * (1.0f - alpha);

  float s = o[0] + o[1] + o[2] + o[3];
  #pragma unroll
  for (int m = 16; m >= 1; m >>= 1) s += __shfl_xor(s, m, 32);
  const float mu = s * (1.0f / FEAT);

  float vs = 0.f;
  #pragma unroll
  for (int i = 0; i < 4; ++i) { const float dlt = o[i] - mu; vs += dlt * dlt; }
  #pragma unroll
  for (int m = 16; m >= 1; m >>= 1) vs += __shfl_xor(vs, m, 32);
  const float rinv = rsqrtf(vs * (1.0f / FEAT) + LN_EPS);

  const float* g = ln_g + (size_t)t * FEAT + lane * 4;
  const float* b = ln_b + (size_t)t * FEAT + lane * 4;
  float4 res;
  res.x = (o[0] - mu) * rinv * g[0] + b[0];
  res.y = (o[1] - mu) * rinv * g[1] + b[1];
  res.z = (o[2] - mu) * rinv * g[2] + b[2];
  res.w = (o[3] - mu) * rinv * g[3] + b[3];
  ((float4*)(out + (size_t)node * FEAT))[lane] = res;
}

// ---------------------------------------------------------------------------
extern "C" void kernel_launch(void* const* d_in, const int* in_sizes, int n_in,
                              void* d_out, int out_size, void* d_ws, size_t ws_size,
                              hipStream_t stream)
{
  const float* x       = (const float*)d_in[0];
  const int*   ntype   = (const int*)  d_in[1];
  const int*   src     = (const int*)  d_in[2];
  const int*   dst     = (const int*)  d_in[3];
  const int*   etype   = (const int*)  d_in[4];
  const float* Wk      = (const float*)d_in[5];
  const float* bk      = (const float*)d_in[6];
  const float* Wq      = (const float*)d_in[7];
  const float* bq      = (const float*)d_in[8];
  const float* Wv      = (const float*)d_in[9];
  const float* bv      = (const float*)d_in[10];
  const float* Wa      = (const float*)d_in[11];
  const float* ba      = (const float*)d_in[12];
  const float* rel_pri = (const float*)d_in[13];
  const float* rel_att = (const float*)d_in[14];
  const float* rel_msg = (const float*)d_in[15];
  const float* skip    = (const float*)d_in[16];
  const float* ln_g    = (const float*)d_in[17];
  const float* ln_b    = (const float*)d_in[18];
  float* out = (float*)d_out;

  float* ws = (float*)d_ws;
  const size_t nf   = (size_t)N_NODES * FEAT;   // 6.4M floats
  const size_t oK = 0;
  const size_t oQ = oK + nf;
  const size_t oV = oQ + nf;
  const size_t oT = oV + nf;                    // numerator / t buffer
  const size_t oA = oT + nf;                    // [E,H] logits -> probs
  const size_t oM = oA + (size_t)N_EDGES * HH;  // [NR,H] segment max
  const size_t oD = oM + (size_t)NRSEG * HH;    // [NR,H] segment denom
  const size_t oC = oD + (size_t)NRSEG * HH;    // [NR] counts (int)
  float* K    = ws + oK;   // reused as TRANS after edge phase
  float* Q    = ws + oQ;
  float* V    = ws + oV;
  float* TN   = ws + oT;
  float* Abuf = ws + oA;
  float* Mseg = ws + oM;
  float* Dseg = ws + oD;
  int*   Cnt  = (int*)(ws + oC);

  // 1. init segment state + accumulator
  {
    const int total = N_NODES * FEAT;
    init_ws<<<(total + 255) / 256, 256, 0, stream>>>(Mseg, Dseg, Cnt, TN);
  }

  // 2-4. WMMA typed projections
  dim3 pgrid(N_NODES / 16, FEAT / 16);
  typed_proj_wmma<<<pgrid, 32, 0, stream>>>(x, ntype, Wk, bk, K);
  typed_proj_wmma<<<pgrid, 32, 0, stream>>>(x, ntype, Wq, bq, Q);
  typed_proj_wmma<<<pgrid, 32, 0, stream>>>(x, ntype, Wv, bv, V);

  // 5-7. edge phase
  const int eblocks = (N_EDGES * HH + 255) / 256;
  edge_logits<<<eblocks, 256, 0, stream>>>(K, Q, src, dst, etype, rel_att,
                                           rel_pri, Abuf, Mseg, Cnt);
  edge_exp<<<eblocks, 256, 0, stream>>>(dst, etype, Abuf, Mseg, Dseg);
  edge_agg<<<eblocks, 256, 0, stream>>>(V, src, dst, etype, rel_msg, Abuf,
                                        Dseg, TN);

  // 8. mean over active etypes
  const int nblocks = (N_NODES * 32 + 255) / 256;
  node_scale<<<nblocks, 256, 0, stream>>>(TN, Cnt);

  // 9. output typed projection (TRANS reuses K's buffer; K is dead now)
  typed_proj_wmma<<<pgrid, 32, 0, stream>>>(TN, ntype, Wa, ba, K);

  // 10. skip-gate + per-type LayerNorm
  node_final<<<nblocks, 256, 0, stream>>>(K, x, ntype, skip, ln_g, ln_b, out);
}